// Discriminator_8744553415337
// MI455X (gfx1250) — compile-verified
//
#include <hip/hip_runtime.h>
#include <hip/hip_bf16.h>
#include <math.h>

typedef __attribute__((ext_vector_type(2))) float v2f;
typedef __attribute__((ext_vector_type(8))) float v8f;

#define DIM 64
#define PAIR_STRIDE 80   // pairs per K-pair row: 64 cols + 16 pad -> conflict-free halves

// ---------------- bucket sort by relation (8 relations) ----------------

__global__ void disc_zero_kernel(int* hist, int* cursor, int nrel) {
    int t = threadIdx.x;
    if (t < nrel) { hist[t] = 0; cursor[t] = 0; }
}

__global__ void disc_hist_kernel(const int* __restrict__ rel, int B, int* __restrict__ hist) {
    int b = blockIdx.x * blockDim.x + threadIdx.x;
    if (b < B) atomicAdd(&hist[rel[b]], 1);
}

__global__ void disc_scan_kernel(const int* __restrict__ hist, int* __restrict__ offsets,
                                 int* __restrict__ cursor, int nrel) {
    if (threadIdx.x == 0 && blockIdx.x == 0) {
        int acc = 0;
        for (int r = 0; r < nrel; ++r) {
            offsets[r] = acc;
            cursor[r]  = acc;
            acc += hist[r];
        }
    }
}

__global__ void disc_scatter_kernel(const int* __restrict__ rel, int B,
                                    int* __restrict__ cursor, int* __restrict__ perm) {
    int b = blockIdx.x * blockDim.x + threadIdx.x;
    if (b < B) {
        int p = atomicAdd(&cursor[rel[b]], 1);
        perm[p] = b;
    }
}

// ---------------- main scoring kernel: fp32 WMMA 16x16x4 ----------------
// grid = (ceil(B/128), nrel), block = 256 (8 waves). Each wave handles a
// 16-row tile of bucket r: T(16x64) = E(16x64) @ R(64x64), then row-dot with
// the gathered neighbor rows, sigmoid, scatter-store.
//
// LDS layout for R: K-pair interleaved. Pair p (rows 2p,2p+1), column c:
//   Rs[(p*PAIR_STRIDE + c)*2 + (row&1)]
// so a B fragment (rows krow,krow+1 at column c) is one aligned ds_load_b64.
//
// All gathered table accesses use 32-bit element offsets (max ~6.4M floats)
// so the backend can use GVS addressing (SGPR base + 32-bit VGPR offset)
// instead of per-lane 64-bit address arithmetic.

__global__ __launch_bounds__(256) void disc_score_kernel(
    const int* __restrict__ node_idx,
    const int* __restrict__ nb_idx,
    const float* __restrict__ node_table,
    const float* __restrict__ rel_table,
    const int* __restrict__ perm,
    const int* __restrict__ offsets,
    const int* __restrict__ counts,
    float* __restrict__ out)
{
    __shared__ __align__(16) float Rs[(DIM / 2) * PAIR_STRIDE * 2];  // 20 KB

    const int r      = blockIdx.y;
    const int count  = counts[r];
    const int start  = offsets[r];
    const int bbase  = blockIdx.x * 128;
    if (bbase >= count) return;                 // uniform across block: safe pre-barrier exit

    // stage R (64x64 f32) into interleaved LDS: each thread handles 2 pair-chunks
    const float4* Rg = (const float4*)(rel_table + (size_t)r * (DIM * DIM));
    #pragma unroll
    for (int j = 0; j < 2; ++j) {
        int pidx  = threadIdx.x + j * 256;          // 0..511: (pair p, 4-col chunk)
        int p     = pidx >> 4;                      // 0..31
        int c4    = (pidx & 15) << 2;               // 0,4,...,60
        float4 lo = Rg[(2 * p)     * 16 + (c4 >> 2)];   // R[2p  ][c4..c4+3]
        float4 hi = Rg[(2 * p + 1) * 16 + (c4 >> 2)];   // R[2p+1][c4..c4+3]
        float* dst = &Rs[(p * PAIR_STRIDE + c4) * 2];
        ((v2f*)dst)[0] = (v2f){lo.x, hi.x};
        ((v2f*)dst)[1] = (v2f){lo.y, hi.y};
        ((v2f*)dst)[2] = (v2f){lo.z, hi.z};
        ((v2f*)dst)[3] = (v2f){lo.w, hi.w};
    }
    __syncthreads();

    const int wave  = threadIdx.x >> 5;
    const int lane  = threadIdx.x & 31;
    const int half  = lane >> 4;                // 0: lanes 0-15, 1: lanes 16-31
    const int l16   = lane & 15;
    const int tbase = bbase + wave * 16;
    if (tbase >= count) return;                 // post-barrier wave exit: safe

    // A-row gather for this lane (row m = l16 of the tile); clamp partial tiles
    int posA = tbase + l16;
    int pA   = perm[start + min(posA, count - 1)];
    const int aoff = node_idx[pA] * DIM;        // 32-bit element offset

    v8f acc[4];
    #pragma unroll
    for (int n = 0; n < 4; ++n)
        acc[n] = (v8f){0.f, 0.f, 0.f, 0.f, 0.f, 0.f, 0.f, 0.f};

    // B-frag base: pair index = 2k + half, so float index =
    //   ((2k+half)*PAIR_STRIDE + 16n + l16)*2 = base + k*4*PAIR_STRIDE + n*32
    const float* Bp = &Rs[(half * PAIR_STRIDE + l16) * 2];

    // T = E @ R : 16 K-steps x 4 N-tiles of v_wmma_f32_16x16x4_f32
    #pragma unroll
    for (int k = 0; k < 16; ++k) {
        const int krow = 4 * k + 2 * half;
        v2f a = *(const v2f*)(node_table + (aoff + krow));  // A frag: E[m, krow..krow+1]
        #pragma unroll
        for (int n = 0; n < 4; ++n) {
            v2f b = *(const v2f*)(Bp + k * (4 * PAIR_STRIDE) + n * 32);  // ds_load_b64
            acc[n] = __builtin_amdgcn_wmma_f32_16x16x4_f32(
                false, a, false, b, (short)0, acc[n], false, false);
        }
    }

    // neighbor row offsets for the 8 rows this lane touches (row = v + 8*half)
    int nbo[8];
    #pragma unroll
    for (int v = 0; v < 8; ++v) {
        int pos = tbase + v + 8 * half;
        int p   = perm[start + min(pos, count - 1)];
        nbo[v]  = nb_idx[p] * DIM;              // 32-bit element offset
    }

    // s[v] = partial_c T[row, c] * N[row, c] over this lane's 4 columns
    float s[8];
    #pragma unroll
    for (int v = 0; v < 8; ++v) s[v] = 0.f;
    #pragma unroll
    for (int n = 0; n < 4; ++n) {
        int c = 16 * n + l16;
        #pragma unroll
        for (int v = 0; v < 8; ++v) {
            float nv = node_table[nbo[v] + c];
            s[v] = fmaf(acc[n][v], nv, s[v]);
        }
    }

    // reduce the 16 column-lanes of each half-wave
    #pragma unroll
    for (int m = 8; m >= 1; m >>= 1) {
        #pragma unroll
        for (int v = 0; v < 8; ++v)
            s[v] += __shfl_xor(s[v], m, 32);
    }

    if (l16 == 0) {
        #pragma unroll
        for (int v = 0; v < 8; ++v) {
            int pos = tbase + v + 8 * half;
            if (pos < count) {
                float sc = s[v];
                out[perm[start + pos]] = 1.0f / (1.0f + expf(-sc));
            }
        }
    }
}

// ---------------- launch ----------------

extern "C" void kernel_launch(void* const* d_in, const int* in_sizes, int n_in,
                              void* d_out, int out_size, void* d_ws, size_t ws_size,
                              hipStream_t stream) {
    const int*   node_idx          = (const int*)d_in[0];
    const int*   relation_idx      = (const int*)d_in[1];
    const int*   node_neighbor_idx = (const int*)d_in[2];
    const float* node_table        = (const float*)d_in[3];
    const float* relation_table    = (const float*)d_in[4];
    float*       out               = (float*)d_out;

    const int B    = in_sizes[0];
    const int nrel = in_sizes[4] / (DIM * DIM);   // = 8

    // workspace layout (ints): hist[64] | offsets[64] | cursor[64] | perm[B]
    int* hist    = (int*)d_ws;
    int* offsets = hist + 64;
    int* cursor  = hist + 128;
    int* perm    = hist + 192;

    disc_zero_kernel<<<1, 64, 0, stream>>>(hist, cursor, nrel);
    disc_hist_kernel<<<(B + 255) / 256, 256, 0, stream>>>(relation_idx, B, hist);
    disc_scan_kernel<<<1, 32, 0, stream>>>(hist, offsets, cursor, nrel);
    disc_scatter_kernel<<<(B + 255) / 256, 256, 0, stream>>>(relation_idx, B, cursor, perm);

    dim3 grid((B + 127) / 128, nrel);
    disc_score_kernel<<<grid, 256, 0, stream>>>(
        node_idx, node_neighbor_idx, node_table, relation_table,
        perm, offsets, hist, out);
}